// ReweightingModule_33397665694032
// MI455X (gfx1250) — compile-verified
//
#include <hip/hip_runtime.h>

#define C64 64

typedef __attribute__((ext_vector_type(2))) float v2f;
typedef __attribute__((ext_vector_type(8))) float v8f;

// ---------------------------------------------------------------------------
// Kernel 1: per radius-edge weight w[e] = sigmoid(-||R0[d]p_s + t0[d] - x_s||^2 + 0.01)
// fused with m1 (one-hot first layer collapses to per-cluster max of w).
// One wave32 per destination node. Channel c lives at position c; lane owns
// channels {2*lane, 2*lane+1} for vectorized IO.
// ---------------------------------------------------------------------------
__global__ void edge_w_m1_kernel(const float* __restrict__ x0,
                                 const float* __restrict__ pos0,
                                 const float* __restrict__ R0,
                                 const float* __restrict__ t0,
                                 const int* __restrict__ assign01,
                                 const int* __restrict__ rsrc,
                                 float* __restrict__ w_out,
                                 float* __restrict__ m1,
                                 float* __restrict__ s1,
                                 int N, int K) {
  __shared__ int lmax[8][C64];
  const int lane = threadIdx.x & 31;
  const int wv = threadIdx.x >> 5;
  const int d = blockIdx.x * 8 + wv;
  const bool act = d < N;
  lmax[wv][2 * lane]     = 0;
  lmax[wv][2 * lane + 1] = 0;
  __syncthreads();
  if (act) {
    const size_t rb = (size_t)d * 9, tb = (size_t)d * 3;
    const float r00 = R0[rb+0], r01 = R0[rb+1], r02 = R0[rb+2];
    const float r10 = R0[rb+3], r11 = R0[rb+4], r12 = R0[rb+5];
    const float r20 = R0[rb+6], r21 = R0[rb+7], r22 = R0[rb+8];
    const float tx = t0[tb+0], ty = t0[tb+1], tz = t0[tb+2];
    for (int l = lane; l < K; l += 32) {
      const size_t e = (size_t)d * K + l;
      const int s = rsrc[e];
      const size_t sb = (size_t)s * 3;
      const float px = pos0[sb+0], py = pos0[sb+1], pz = pos0[sb+2];
      const float xx = x0[sb+0],  xy = x0[sb+1],  xz = x0[sb+2];
      const float e0 = r00*px + r01*py + r02*pz + tx - xx;
      const float e1 = r10*px + r11*py + r12*pz + ty - xy;
      const float e2 = r20*px + r21*py + r22*pz + tz - xz;
      const float d2 = e0*e0 + e1*e1 + e2*e2;
      const float w = 1.0f / (1.0f + __expf(d2 - 0.01f));
      w_out[e] = w;
      const int cl = assign01[s] & (C64 - 1);
      atomicMax(&lmax[wv][cl], __float_as_int(w));   // w > 0: int order == float order
    }
  }
  __syncthreads();
  if (act) {
    const float v0 = __int_as_float(lmax[wv][2 * lane]);
    const float v1 = __int_as_float(lmax[wv][2 * lane + 1]);
    *(float2*)(m1 + (size_t)d * C64 + 2 * lane) = make_float2(v0, v1);
    float su = v0 + v1;
    for (int off = 16; off > 0; off >>= 1) su += __shfl_xor(su, off, 32);
    if (lane == 0) s1[d] = su;
  }
}

// ---------------------------------------------------------------------------
// Kernel 2: max-aggregation layer: out[d][c] = max_e w[e]*in[src[e]][c].
// One wave32 per dst node; lane covers channels {2*lane, 2*lane+1} (b64 loads);
// (src,w) broadcast to SGPRs via readlane -> scalar-base addressing.
// KT: compile-time K (0 = runtime).
// ---------------------------------------------------------------------------
template<int KT>
__global__ void max_prop_kernel(const float* __restrict__ fin,
                                const float* __restrict__ w,
                                const int* __restrict__ rsrc,
                                float* __restrict__ fout,
                                float* __restrict__ ssum,
                                int N, int Krt) {
  const int K = KT ? KT : Krt;
  const int lane = threadIdx.x & 31;
  const int d = blockIdx.x * 8 + (threadIdx.x >> 5);
  if (d >= N) return;
  const unsigned ch = 2u * lane;
  float acc0 = 0.0f, acc1 = 0.0f;
  for (int base = 0; base < K; base += 32) {
    const int cnt = (K - base < 32) ? (K - base) : 32;
    int sl = 0, wl = 0;
    if (lane < cnt) {
      const size_t e = (size_t)d * K + base + lane;
      sl = rsrc[e];
      wl = __float_as_int(w[e]);
    }
#pragma unroll
    for (int i = 0; i < (KT ? KT : 1); ++i) {
      if (KT ? (i >= cnt) : true) break;          // constexpr-unrolled when KT
      const int s = __builtin_amdgcn_readlane(sl, i);
      const float wv = __int_as_float(__builtin_amdgcn_readlane(wl, i));
      const float2 f = *(const float2*)(fin + (unsigned)s * C64 + ch);
      acc0 = fmaxf(acc0, wv * f.x);
      acc1 = fmaxf(acc1, wv * f.y);
    }
    if (!KT) {                                    // runtime-K fallback
      for (int i = 0; i < cnt; ++i) {
        const int s = __builtin_amdgcn_readlane(sl, i);
        const float wv = __int_as_float(__builtin_amdgcn_readlane(wl, i));
        const float2 f = *(const float2*)(fin + (unsigned)s * C64 + ch);
        acc0 = fmaxf(acc0, wv * f.x);
        acc1 = fmaxf(acc1, wv * f.y);
      }
    }
  }
  *(float2*)(fout + (size_t)d * C64 + ch) = make_float2(acc0, acc1);
  float su = acc0 + acc1;
  for (int off = 16; off > 0; off >>= 1) su += __shfl_xor(su, off, 32);
  if (lane == 0) ssum[d] = su;
}

// ---------------------------------------------------------------------------
// Kernel 3: conf[n] = sigmoid(s1 + s2 + 3*s3 - 17)   (m3 == m4 == m5)
// ---------------------------------------------------------------------------
__global__ void conf_kernel(const float* __restrict__ s1,
                            const float* __restrict__ s2,
                            const float* __restrict__ s3,
                            float* __restrict__ conf, int N) {
  const int n = blockIdx.x * blockDim.x + threadIdx.x;
  if (n >= N) return;
  const float t = 17.0f - (s1[n] + s2[n] + 3.0f * s3[n]);
  conf[n] = 1.0f / (1.0f + __expf(t));
}

// ---------------------------------------------------------------------------
// Kernel 3b: padded value table Vtab[i][0:16] = [R0[i](9), t0[i](3), 1, 0,0,0]
// ---------------------------------------------------------------------------
__global__ void vtab_kernel(const float* __restrict__ R0,
                            const float* __restrict__ t0,
                            float* __restrict__ Vtab, int N) {
  const long i = (long)blockIdx.x * blockDim.x + threadIdx.x;
  if (i >= (long)N * 16) return;
  const int s = (int)(i >> 4);
  const int j = (int)(i & 15);
  float v;
  if (j < 9)       v = R0[(size_t)s * 9 + j];
  else if (j < 12) v = t0[(size_t)s * 3 + (j - 9)];
  else             v = (j == 12) ? 1.0f : 0.0f;
  Vtab[i] = v;
}

// ---------------------------------------------------------------------------
// Kernel 3c: packed per-knn-edge record {src, conf[src]} as one 64-bit word;
// entries in [E2, Epad) are zero-filled so the WMMA kernel needs no bounds
// checks (zero conf => zero A-column => no contribution; src=0 => safe gather).
// ---------------------------------------------------------------------------
__global__ void edge_sw_kernel(const int* __restrict__ ksrc,
                               const float* __restrict__ conf,
                               unsigned long long* __restrict__ sw,
                               long E2, long Epad) {
  const long e = (long)blockIdx.x * blockDim.x + threadIdx.x;
  if (e >= Epad) return;
  unsigned long long q = 0ull;
  if (e < E2) {
    const int s = ksrc[e];
    q = (unsigned long long)(unsigned)s |
        ((unsigned long long)(unsigned)__float_as_int(conf[s]) << 32);
  }
  sw[e] = q;
}

// ---------------------------------------------------------------------------
// Kernel 4: KNN weighted average as block-sparse GEMM on V_WMMA_F32_16X16X4_F32.
// One wave32 per tile of 16 dst nodes. Iterate dst node m = 0..15; node m's
// EPD edges are processed in ceil(EPD/4) chunks of K=4; the owner is the loop
// index m (no division). sw is padded, so no runtime bounds checks: only the
// compile-time j<EPD predicate folds into the A-operand cndmask. The packed
// (src,conf) record is one b64 load whose low half feeds the B gather address
// unconditionally (load cannot be predicated). Column 12 accumulates the
// denominator (Vtab[s][12]==1).
// A/B VGPR layout (ISA 7.12.2): lanes 0-15 <-> K={0,1}, lanes 16-31 <-> K={2,3}.
// EPDT: compile-time edges-per-dst (0 = runtime).
// ---------------------------------------------------------------------------
template<int EPDT>
__global__ void wavg_wmma_kernel(const float* __restrict__ Vtab,
                                 const unsigned long long* __restrict__ sw,
                                 float* __restrict__ Rout,
                                 float* __restrict__ tout,
                                 int N, int EPDrt) {
  const int EPD = EPDT ? EPDT : EPDrt;
  const int lane = threadIdx.x & 31;
  const int half = lane >> 4;           // 0: K={0,1}, 1: K={2,3}
  const int n = lane & 15;              // B column / A row (M)
  const int tiles = (N + 15) >> 4;
  const int tile = blockIdx.x * (blockDim.x >> 5) + (threadIdx.x >> 5);
  if (tile >= tiles) return;            // uniform per wave: EXEC stays all-ones
  const unsigned ebase = (unsigned)tile * 16u * (unsigned)EPD;
  const int kk = 2 * half;
  v8f acc = {0.f, 0.f, 0.f, 0.f, 0.f, 0.f, 0.f, 0.f};
#pragma unroll 4
  for (int m = 0; m < 16; ++m) {
    const bool own = (n == m);
    const unsigned nb = ebase + (unsigned)m * (unsigned)EPD + (unsigned)kk;
#pragma unroll
    for (int cc = 0; cc < (EPDT ? ((EPDT + 3) >> 2) : 1); ++cc) {
      if (!EPDT) break;                 // compile-time path only
      const int j0 = 4 * cc + kk;       // constexpr after unroll
      const bool vj0 = j0 < EPD;        // constexpr
      const bool vj1 = j0 + 1 < EPD;    // constexpr
      const unsigned e0 = nb + (unsigned)(vj0 ? 4 * cc : -kk);
      const unsigned e1 = nb + (unsigned)(vj1 ? 4 * cc + 1 : -kk);
      const unsigned long long q0 = sw[e0];
      const unsigned long long q1 = sw[e1];
      const unsigned s0 = (unsigned)q0;
      const unsigned s1 = (unsigned)q1;
      const float w0 = __int_as_float((int)(q0 >> 32));
      const float w1 = __int_as_float((int)(q1 >> 32));
      v2f a, b;
      a.x = (own && vj0) ? w0 : 0.0f;
      a.y = (own && vj1) ? w1 : 0.0f;
      b.x = Vtab[(s0 << 4) + n];
      b.y = Vtab[(s1 << 4) + n];
      acc = __builtin_amdgcn_wmma_f32_16x16x4_f32(
          false, a, false, b, (short)0, acc, false, false);
    }
    if (!EPDT) {                        // runtime-EPD fallback (same structure)
      const int CPN = (EPD + 3) >> 2;
      for (int cc = 0; cc < CPN; ++cc) {
        const int j0 = 4 * cc + kk;
        const bool vj0 = j0 < EPD;
        const bool vj1 = j0 + 1 < EPD;
        const unsigned e0 = nb + (unsigned)(vj0 ? 4 * cc : -kk);
        const unsigned e1 = nb + (unsigned)(vj1 ? 4 * cc + 1 : -kk);
        const unsigned long long q0 = sw[e0];
        const unsigned long long q1 = sw[e1];
        const unsigned s0 = (unsigned)q0;
        const unsigned s1 = (unsigned)q1;
        const float w0 = __int_as_float((int)(q0 >> 32));
        const float w1 = __int_as_float((int)(q1 >> 32));
        v2f a, b;
        a.x = (own && vj0) ? w0 : 0.0f;
        a.y = (own && vj1) ? w1 : 0.0f;
        b.x = Vtab[(s0 << 4) + n];
        b.y = Vtab[(s1 << 4) + n];
        acc = __builtin_amdgcn_wmma_f32_16x16x4_f32(
            false, a, false, b, (short)0, acc, false, false);
      }
    }
  }
#pragma unroll
  for (int r = 0; r < 8; ++r) {
    // denominator = column 12: lane 12 (rows 0..7) / lane 28 (rows 8..15)
    const float den = __shfl(acc[r], half ? 28 : 12, 32) + 1e-8f;
    const float val = acc[r] / den;
    const int node = tile * 16 + r + half * 8;
    if (node < N) {
      if (n < 9)       Rout[(size_t)node * 9 + n] = val;
      else if (n < 12) tout[(size_t)node * 3 + (n - 9)] = val;
    }
  }
}

// ---------------------------------------------------------------------------
// Kernel 5: x_out[n] = R_out[n] @ pos0[n] + t_out[n]
// ---------------------------------------------------------------------------
__global__ void xout_kernel(const float* __restrict__ Rout,
                            const float* __restrict__ tout,
                            const float* __restrict__ pos0,
                            float* __restrict__ xout, int N) {
  const int i = blockIdx.x * blockDim.x + threadIdx.x;
  if (i >= N) return;
  const float* R = Rout + (size_t)i * 9;
  const float px = pos0[(size_t)i*3+0], py = pos0[(size_t)i*3+1], pz = pos0[(size_t)i*3+2];
  xout[(size_t)i*3+0] = R[0]*px + R[1]*py + R[2]*pz + tout[(size_t)i*3+0];
  xout[(size_t)i*3+1] = R[3]*px + R[4]*py + R[5]*pz + tout[(size_t)i*3+1];
  xout[(size_t)i*3+2] = R[6]*px + R[7]*py + R[8]*pz + tout[(size_t)i*3+2];
}

extern "C" void kernel_launch(void* const* d_in, const int* in_sizes, int n_in,
                              void* d_out, int out_size, void* d_ws, size_t ws_size,
                              hipStream_t stream) {
  const float* x0       = (const float*)d_in[0];
  const float* pos0     = (const float*)d_in[1];
  const float* R0       = (const float*)d_in[2];
  const float* t0       = (const float*)d_in[3];
  const int*   assign01 = (const int*)d_in[4];
  const int*   rsrc     = (const int*)d_in[5];
  // d_in[6] radius_dst: edges grouped by dst (repeat(arange(N),K)) -> dst = e/K
  const int*   ksrc     = (const int*)d_in[7];
  // d_in[8] knn_dst: same grouping

  const int N  = in_sizes[0] / 3;
  const int KR = in_sizes[5] / N;     // 32
  const int KK = in_sizes[7] / N;     // 10
  const long E2 = (long)N * KK;
  const int tiles = (N + 15) / 16;
  const long Epad = (long)tiles * 16 * KK + 4;   // padded edge table length

  float* ws = (float*)d_ws;
  size_t off = 0;
  float* w    = ws + off; off += (size_t)N * KR;
  float* mA   = ws + off; off += (size_t)N * C64;
  float* mB   = ws + off; off += (size_t)N * C64;
  float* s1   = ws + off; off += N;
  float* s2   = ws + off; off += N;
  float* s3   = ws + off; off += N;
  float* conf = ws + off; off += N;
  float* Vtab = ws + off; off += (size_t)N * 16;
  off = (off + 1) & ~(size_t)1;                   // 8B alignment
  unsigned long long* sw = (unsigned long long*)(ws + off); off += 2 * (size_t)Epad;

  float* xout = (float*)d_out;                  // [N,3]
  float* Rout = xout + (size_t)N * 3;           // [N,3,3]
  float* tout = xout + (size_t)N * 12;          // [N,3]

  const int nodeBlocks = (N + 7) / 8;           // 8 waves (8 dst nodes) per block
  edge_w_m1_kernel<<<nodeBlocks, 256, 0, stream>>>(x0, pos0, R0, t0, assign01,
                                                   rsrc, w, mA, s1, N, KR);
  if (KR == 32) {
    max_prop_kernel<32><<<nodeBlocks, 256, 0, stream>>>(mA, w, rsrc, mB, s2, N, KR);
    max_prop_kernel<32><<<nodeBlocks, 256, 0, stream>>>(mB, w, rsrc, mA, s3, N, KR);
  } else {
    max_prop_kernel<0><<<nodeBlocks, 256, 0, stream>>>(mA, w, rsrc, mB, s2, N, KR);
    max_prop_kernel<0><<<nodeBlocks, 256, 0, stream>>>(mB, w, rsrc, mA, s3, N, KR);
  }
  conf_kernel<<<(N + 255) / 256, 256, 0, stream>>>(s1, s2, s3, conf, N);
  vtab_kernel<<<(int)(((long)N * 16 + 255) / 256), 256, 0, stream>>>(R0, t0, Vtab, N);
  edge_sw_kernel<<<(int)((Epad + 255) / 256), 256, 0, stream>>>(ksrc, conf, sw, E2, Epad);
  if (KK == 10) {
    wavg_wmma_kernel<10><<<(tiles + 3) / 4, 128, 0, stream>>>(Vtab, sw, Rout, tout, N, KK);
  } else {
    wavg_wmma_kernel<0><<<(tiles + 3) / 4, 128, 0, stream>>>(Vtab, sw, Rout, tout, N, KK);
  }
  xout_kernel<<<(N + 255) / 256, 256, 0, stream>>>(Rout, tout, pos0, xout, N);
}